// GATClassifier_20229295964953
// MI455X (gfx1250) — compile-verified
//
#include <hip/hip_runtime.h>
#include <hip/hip_bf16.h>

// ---------------------------------------------------------------------------
// GATv2 (2 layers, heads=2 then 1) + linear head for gfx1250 (CDNA5, wave32).
// Dense transforms use V_WMMA_F32_16X16X4_F32 (full-fp32 matrix core path).
// Edge phase is wave-per-edge with L2-resident atomics (outputs fit in 192MB L2).
// ---------------------------------------------------------------------------

typedef __attribute__((ext_vector_type(2))) float v2f;
typedef __attribute__((ext_vector_type(8))) float v8f;

// ---- ordered-uint encoding so unsigned atomicMax implements float max ------
__device__ __forceinline__ unsigned ord_f32(float f) {
    unsigned u = __float_as_uint(f);
    return (u & 0x80000000u) ? ~u : (u | 0x80000000u);
}
__device__ __forceinline__ float deord_f32(unsigned e) {
    unsigned u = (e & 0x80000000u) ? (e ^ 0x80000000u) : ~e;
    return __uint_as_float(u);
}

// ---------------------------------------------------------------------------
// fp32 WMMA GEMM: C[M,N] = A[M,K] @ B[K,N] (+bias).  One wave computes NT
// adjacent 16x16 tiles (shared A-fragment).  blockDim = (32, GEMM_WPB).
// Requires M % 16 == 0 (50000 = 3125*16).  Out-of-range N columns are
// CLAMPED at load (column n of D depends only on column n of B, so garbage
// in clamped columns is never stored) -> loads are unconditional and EXEC
// stays full.  B is addressed as B[int_offset] directly off the kernel arg
// so address-space inference keeps the loads on the GLOBAL path (loop-carried
// pointers regressed them to FLAT in a previous build).
//
// A-frag (16x4 f32): lane = (khalf<<4)|m, v2f = {K=kh, K=kh+1}, kh = 0|2.
// B-frag (4x16 f32): lanes 0-15 rows K=0,1; lanes 16-31 rows K=2,3.
// C/D: VGPR r -> row m0 + r (+8 for lanes 16-31), col = lane%16.
// ---------------------------------------------------------------------------
#define GEMM_WPB 8
template <int NT>
__global__ void wmma_gemm_f32(const float* __restrict__ A,
                              const float* __restrict__ B,
                              float* __restrict__ C,
                              const float* __restrict__ bias,
                              int M, int N, int K, int do_relu) {
    const int tileM = blockIdx.x * GEMM_WPB + threadIdx.y;   // wave-uniform
    if (tileM * 16 >= M) return;                              // uniform per wave

    const int lane  = threadIdx.x;           // 0..31
    const int lm    = lane & 15;             // row (A) / col (B,C)
    const int khalf = (lane >> 4) << 1;      // 0 or 2
    const int m0 = tileM * 16;
    const int n0 = blockIdx.y * NT * 16;

    int col[NT];
    int boff[NT];                            // 32-bit offsets into B (global)
    v8f acc[NT];
    #pragma unroll
    for (int t = 0; t < NT; ++t) {
        col[t] = n0 + t * 16 + lm;
        const int colc = col[t] < N ? col[t] : N - 1;        // clamp, not mask
        boff[t] = khalf * N + colc;
        acc[t] = v8f{0.f, 0.f, 0.f, 0.f, 0.f, 0.f, 0.f, 0.f};
    }

    const float* __restrict__ arow = A + (size_t)(m0 + lm) * K + khalf;
    const int bstep = 4 * N;

    #pragma unroll 4
    for (int k = 0; k < K; k += 4) {
        const v2f a = *(const v2f*)(arow + k);               // 8B aligned
        #pragma unroll
        for (int t = 0; t < NT; ++t) {
            v2f b;
            b.x = B[boff[t]];
            b.y = B[boff[t] + N];
            boff[t] += bstep;
            acc[t] = __builtin_amdgcn_wmma_f32_16x16x4_f32(
                         /*neg_a=*/false, a, /*neg_b=*/false, b,
                         /*c_mod=*/(short)0, acc[t],
                         /*reuse_a=*/false, /*reuse_b=*/false);
        }
    }

    const int rbase = m0 + ((lane >> 4) << 3);               // +8 for upper half
    #pragma unroll
    for (int t = 0; t < NT; ++t) {
        if (col[t] < N) {
            const float bb = bias ? bias[col[t]] : 0.f;
            #pragma unroll
            for (int r = 0; r < 8; ++r) {
                float v = acc[t][r] + bb;
                if (do_relu) v = v > 0.f ? v : 0.f;
                C[(size_t)(rbase + r) * N + col[t]] = v;
            }
        }
    }
}

// ---------------------------------------------------------------------------
// Edge kernel 1: per-edge attention logits + segment max (ordered atomicMax).
// One wave per edge; lane owns HC/32 contiguous channels.
// logit[h] = sum_c att[h*C+c] * leaky_relu(xl[src][h,c] + xr[dst][h,c], 0.2)
// ---------------------------------------------------------------------------
__global__ void edge_logits(const float* __restrict__ xl,
                            const float* __restrict__ xr,
                            const long long* __restrict__ ei,
                            const float* __restrict__ att,
                            float* __restrict__ logits,
                            unsigned* __restrict__ segmax,
                            int E, int Etot, int H, int C) {
    const int wid  = (int)((blockIdx.x * (size_t)blockDim.x + threadIdx.x) >> 5);
    if (wid >= Etot) return;
    const int lane = threadIdx.x & 31;

    int s, d;
    if (wid < E) { s = (int)ei[wid]; d = (int)ei[(size_t)E + wid]; }
    else         { s = d = wid - E; }                     // self loops

    const int HC  = H * C;
    const int per = HC >> 5;
    const int base = lane * per;
    const float* __restrict__ pl = xl + (size_t)s * HC + base;
    const float* __restrict__ pr = xr + (size_t)d * HC + base;

    float p0 = 0.f, p1 = 0.f;
    for (int j = 0; j < per; ++j) {
        const int c = base + j;
        float v = pl[j] + pr[j];
        v = v > 0.f ? v : 0.2f * v;
        const float av = att[c] * v;
        if (c < C) p0 += av; else p1 += av;
    }
    #pragma unroll
    for (int off = 16; off > 0; off >>= 1) {
        p0 += __shfl_xor(p0, off, 32);
        p1 += __shfl_xor(p1, off, 32);
    }
    if (lane == 0) {
        logits[(size_t)wid * H + 0] = p0;
        atomicMax(&segmax[(size_t)d * H + 0], ord_f32(p0));
        if (H > 1) {
            logits[(size_t)wid * H + 1] = p1;
            atomicMax(&segmax[(size_t)d * H + 1], ord_f32(p1));
        }
    }
}

// ---------------------------------------------------------------------------
// Edge kernel 2: a = exp(logit - max[dst]); segment sum via atomicAdd.
// Thread per (edge, head); overwrites logits with the numerator a.
// ---------------------------------------------------------------------------
__global__ void edge_expsum(const long long* __restrict__ ei,
                            float* __restrict__ logits,
                            const unsigned* __restrict__ segmax,
                            float* __restrict__ segsum,
                            int E, int Etot, int H) {
    const size_t idx = blockIdx.x * (size_t)blockDim.x + threadIdx.x;
    if (idx >= (size_t)Etot * H) return;
    const int e = (int)(idx / H);
    const int h = (int)(idx - (size_t)e * H);
    const int d = (e < E) ? (int)ei[(size_t)E + e] : e - E;
    const float mv = deord_f32(segmax[(size_t)d * H + h]);
    const float a = expf(logits[idx] - mv);
    logits[idx] = a;
    atomicAdd(&segsum[(size_t)d * H + h], a);
}

// ---------------------------------------------------------------------------
// Edge kernel 3: out[dst] += (a / segsum[dst]) * xl[src]; wave per edge.
// ---------------------------------------------------------------------------
__global__ void edge_aggregate(const float* __restrict__ xl,
                               const long long* __restrict__ ei,
                               const float* __restrict__ anum,
                               const float* __restrict__ segsum,
                               float* __restrict__ out,
                               int E, int Etot, int H, int C) {
    const int wid  = (int)((blockIdx.x * (size_t)blockDim.x + threadIdx.x) >> 5);
    if (wid >= Etot) return;
    const int lane = threadIdx.x & 31;

    int s, d;
    if (wid < E) { s = (int)ei[wid]; d = (int)ei[(size_t)E + wid]; }
    else         { s = d = wid - E; }

    const int HC  = H * C;
    const int per = HC >> 5;
    const int base = lane * per;
    const int h = base / C;                  // per-lane constant (per | C)
    const float alpha = anum[(size_t)wid * H + h] / segsum[(size_t)d * H + h];

    const float* __restrict__ pl = xl + (size_t)s * HC + base;
    float* __restrict__ po = out + (size_t)d * HC + base;
    for (int j = 0; j < per; ++j)
        atomicAdd(&po[j], alpha * pl[j]);
}

// ---------------------------------------------------------------------------
__global__ void bias_relu_kernel(float* __restrict__ p,
                                 const float* __restrict__ bias,
                                 size_t n, int hcMask) {
    const size_t i = blockIdx.x * (size_t)blockDim.x + threadIdx.x;
    if (i >= n) return;
    float v = p[i] + bias[i & hcMask];
    p[i] = v > 0.f ? v : 0.f;
}

__global__ void fill_u32(unsigned* __restrict__ p, size_t n, unsigned v) {
    const size_t i = blockIdx.x * (size_t)blockDim.x + threadIdx.x;
    if (i < n) p[i] = v;
}

// ---------------------------------------------------------------------------
extern "C" void kernel_launch(void* const* d_in, const int* in_sizes, int n_in,
                              void* d_out, int out_size, void* d_ws, size_t ws_size,
                              hipStream_t stream) {
    (void)in_sizes; (void)n_in; (void)out_size; (void)ws_size;

    const int NN   = 50000;
    const int E    = 800000;
    const int ETOT = E + NN;          // + self loops
    const int INF  = 128;
    const int HID  = 64;
    const int H1   = 2;
    const int HC1  = H1 * HID;        // 128
    const int NCLS = 10;

    const float*     x    = (const float*)d_in[0];
    const long long* ei   = (const long long*)d_in[1];   // int64 per reference
    const float*     Wl1  = (const float*)d_in[2];
    const float*     Wr1  = (const float*)d_in[3];
    const float*     att1 = (const float*)d_in[4];
    const float*     b1   = (const float*)d_in[5];
    const float*     Wl2  = (const float*)d_in[6];
    const float*     Wr2  = (const float*)d_in[7];
    const float*     att2 = (const float*)d_in[8];
    const float*     b2   = (const float*)d_in[9];
    const float*     Wfc  = (const float*)d_in[10];
    const float*     bfc  = (const float*)d_in[11];
    float*           out  = (float*)d_out;

    // workspace layout (floats); total ~21.1M floats = 84.4 MB
    float*    ws     = (float*)d_ws;
    float*    xl     = ws;                         // 50000*128 = 6.40M
    float*    xr     = ws + 6400000;               // 6.40M
    float*    hbuf   = ws + 12800000;              // 6.40M (h1, later h2)
    float*    logits = ws + 19200000;              // 850000*2 = 1.70M
    unsigned* segmax = (unsigned*)(ws + 20900000); // 100k
    float*    segsum = ws + 21000000;              // 100k

    const dim3 gblk(32, GEMM_WPB);
    const int  mBlocks = ((NN / 16) + GEMM_WPB - 1) / GEMM_WPB;   // 391
    const int  edgeBlocks = (ETOT + 7) / 8;                        // 8 waves/block

    // ============================ Layer 1 =================================
    hipLaunchKernelGGL((wmma_gemm_f32<2>), dim3(mBlocks, HC1 / 32), gblk, 0, stream,
                       x, Wl1, xl, (const float*)nullptr, NN, HC1, INF, 0);
    hipLaunchKernelGGL((wmma_gemm_f32<2>), dim3(mBlocks, HC1 / 32), gblk, 0, stream,
                       x, Wr1, xr, (const float*)nullptr, NN, HC1, INF, 0);

    hipLaunchKernelGGL(fill_u32, dim3((NN * H1 + 255) / 256), dim3(256), 0, stream,
                       segmax, (size_t)NN * H1, 0u);   // 0 < ord(-inf); every node has a self loop
    hipLaunchKernelGGL(fill_u32, dim3((NN * H1 + 255) / 256), dim3(256), 0, stream,
                       (unsigned*)segsum, (size_t)NN * H1, 0u);
    hipLaunchKernelGGL(fill_u32, dim3(((size_t)NN * HC1 + 255) / 256), dim3(256), 0, stream,
                       (unsigned*)hbuf, (size_t)NN * HC1, 0u);

    hipLaunchKernelGGL(edge_logits, dim3(edgeBlocks), dim3(256), 0, stream,
                       xl, xr, ei, att1, logits, segmax, E, ETOT, H1, HID);
    hipLaunchKernelGGL(edge_expsum, dim3(((size_t)ETOT * H1 + 255) / 256), dim3(256), 0, stream,
                       ei, logits, segmax, segsum, E, ETOT, H1);
    hipLaunchKernelGGL(edge_aggregate, dim3(edgeBlocks), dim3(256), 0, stream,
                       xl, ei, logits, segsum, hbuf, E, ETOT, H1, HID);
    hipLaunchKernelGGL(bias_relu_kernel, dim3(((size_t)NN * HC1 + 255) / 256), dim3(256), 0, stream,
                       hbuf, b1, (size_t)NN * HC1, HC1 - 1);

    // ============================ Layer 2 =================================
    hipLaunchKernelGGL((wmma_gemm_f32<2>), dim3(mBlocks, HID / 32), gblk, 0, stream,
                       hbuf, Wl2, xl, (const float*)nullptr, NN, HID, HC1, 0);
    hipLaunchKernelGGL((wmma_gemm_f32<2>), dim3(mBlocks, HID / 32), gblk, 0, stream,
                       hbuf, Wr2, xr, (const float*)nullptr, NN, HID, HC1, 0);

    hipLaunchKernelGGL(fill_u32, dim3((NN + 255) / 256), dim3(256), 0, stream,
                       segmax, (size_t)NN, 0u);
    hipLaunchKernelGGL(fill_u32, dim3((NN + 255) / 256), dim3(256), 0, stream,
                       (unsigned*)segsum, (size_t)NN, 0u);
    hipLaunchKernelGGL(fill_u32, dim3(((size_t)NN * HID + 255) / 256), dim3(256), 0, stream,
                       (unsigned*)hbuf, (size_t)NN * HID, 0u);   // h1 dead after GEMMs

    hipLaunchKernelGGL(edge_logits, dim3(edgeBlocks), dim3(256), 0, stream,
                       xl, xr, ei, att2, logits, segmax, E, ETOT, 1, HID);
    hipLaunchKernelGGL(edge_expsum, dim3(((size_t)ETOT + 255) / 256), dim3(256), 0, stream,
                       ei, logits, segmax, segsum, E, ETOT, 1);
    hipLaunchKernelGGL(edge_aggregate, dim3(edgeBlocks), dim3(256), 0, stream,
                       xl, ei, logits, segsum, hbuf, E, ETOT, 1, HID);
    hipLaunchKernelGGL(bias_relu_kernel, dim3(((size_t)NN * HID + 255) / 256), dim3(256), 0, stream,
                       hbuf, b2, (size_t)NN * HID, HID - 1);

    // ============================ Head ====================================
    hipLaunchKernelGGL((wmma_gemm_f32<1>), dim3(mBlocks, 1), gblk, 0, stream,
                       hbuf, Wfc, out, bfc, NN, NCLS, HID, 0);
}